// GCN_78889959293685
// MI455X (gfx1250) — compile-verified
//
#include <hip/hip_runtime.h>

typedef __attribute__((ext_vector_type(2))) float v2f;
typedef __attribute__((ext_vector_type(8))) float v8f;

#define B_ 16
#define S_ 128
#define H_ 768
#define K_ 4
#define L_ 2

#define MBLK 128
#define NBLK 64
#define KC   32
#define ASTR 36               // floats per A-tile row (KC + 4 pad) -> conflict-free b64 frags
#define KCP  36               // floats per B-tile column (KC + 4 pad) -> conflict-free b64 frags
#define AT   (128 * ASTR)     // 4608 floats = 18 KB per A tile
#define BT   (64 * KCP)       // 2304 floats =  9 KB per B tile (64 cols x KC k-values)

#define ASYNC_WAIT(n) asm volatile("s_wait_asynccnt " #n ::: "memory")

__device__ __forceinline__ v8f wmma4(v2f a, v2f b, v8f c) {
  return __builtin_amdgcn_wmma_f32_16x16x4_f32(false, a, false, b, (short)0, c, false, false);
}

__device__ __forceinline__ unsigned lds_addr(const void* p) {
  return (unsigned)(uintptr_t)p;   // LDS aperture: low 32 bits are the LDS offset
}

// ---- async global->LDS copy streams (per-lane 16B, ASYNCcnt-tracked) ----

// A tile: 128 rows x KC floats from row-major src; lds row-major [m][ASTR]
__device__ __forceinline__ void async_A(const float* gbase, unsigned voff,
                                        unsigned laddr, unsigned gstep) {
#pragma unroll
  for (int p = 0; p < 4; ++p) {
    unsigned l = laddr + (unsigned)p * (32u * ASTR * 4u);
    unsigned g = voff + (unsigned)p * gstep;
    asm volatile("global_load_async_to_lds_b128 %0, %1, %2"
                 :: "v"(l), "v"(g), "s"(gbase) : "memory");
  }
}

// B tile: 64 columns x KC k-floats; src has k contiguous (ld floats between columns);
// lds column-major [c][KCP]
__device__ __forceinline__ void async_B(const float* gbase, unsigned voff,
                                        unsigned laddr, unsigned gstep) {
#pragma unroll
  for (int p = 0; p < 2; ++p) {
    unsigned l = laddr + (unsigned)p * (32u * KCP * 4u);
    unsigned g = voff + (unsigned)p * gstep;
    asm volatile("global_load_async_to_lds_b128 %0, %1, %2"
                 :: "v"(l), "v"(g), "s"(gbase) : "memory");
  }
}

// ---- WMMA on one staged chunk: A row-major [m][ASTR], B column-major [c][KCP] ----
template <int NPROD>
__device__ __forceinline__ void mma_chunk(const float* lA0, const float* lA1,
                                          const float* lB, v8f* acc0, v8f* acc1) {
  int lane = threadIdx.x & 31;
  int wv   = threadIdx.x >> 5;
  int half = lane >> 4;
  int lm   = lane & 15;
  const float* a0p = lA0 + (wv * 16 + lm) * ASTR;
  const float* a1p = lA1 + (wv * 16 + lm) * ASTR;
  const float* bp  = lB + lm * KCP;
#pragma unroll
  for (int kk = 0; kk < KC; kk += 4) {
    int kb = kk + 2 * half;
    v2f a0 = *(const v2f*)(a0p + kb);
    v2f a1 = a0;
    if (NPROD == 2) a1 = *(const v2f*)(a1p + kb);
#pragma unroll
    for (int j = 0; j < 4; ++j) {
      v2f bb = *(const v2f*)(bp + j * 16 * KCP + kb);
      acc0[j] = wmma4(a0, bb, acc0[j]);
      if (NPROD == 2) acc1[j] = wmma4(a1, bb, acc1[j]);
    }
  }
}

// adjT[bk][t][s] = adj[bk][s][t]
__global__ __launch_bounds__(256) void k_transpose_adj(const float* __restrict__ a,
                                                       float* __restrict__ at) {
  __shared__ float tl[32][33];
  int bk = blockIdx.z;
  int x0 = blockIdx.x * 32;
  int y0 = blockIdx.y * 32;
  const float* src = a + (size_t)bk * S_ * S_;
  float* dst = at + (size_t)bk * S_ * S_;
  int tx = threadIdx.x & 31, ty = threadIdx.x >> 5;
#pragma unroll
  for (int i = 0; i < 32; i += 8)
    tl[ty + i][tx] = src[(size_t)(y0 + ty + i) * S_ + x0 + tx];
  __syncthreads();
#pragma unroll
  for (int i = 0; i < 32; i += 8)
    dst[(size_t)(x0 + ty + i) * S_ + y0 + tx] = tl[tx][ty + i];
}

// acc[b] = Wl @ hT[b]^T   (Wl: [H,H] row-major; hsrc: [B,S,H]); out: [B,H,S]
__global__ __launch_bounds__(256) void k_gemm_plain(const float* __restrict__ Wl,
                                                    const float* __restrict__ hsrc,
                                                    float* __restrict__ outp) {
  __shared__ __align__(128) float lds[2 * AT + 2 * BT];
  int b  = blockIdx.x;
  int m0 = blockIdx.y * MBLK;
  int n0 = blockIdx.z * NBLK;
  const float* gA = Wl + (size_t)m0 * H_;
  const float* gB = hsrc + (size_t)b * S_ * H_ + (size_t)n0 * H_;

  unsigned t = threadIdx.x;
  unsigned voff = (((t >> 3) * H_) + (t & 7) * 4) * 4;   // row=t>>3, 4 floats at (t&7)*4
  unsigned lloc = ((t >> 3) * ASTR + (t & 7) * 4) * 4;   // ASTR == KCP
  unsigned base = lds_addr(lds);
  unsigned lA_l = base + lloc;
  unsigned lB_l = base + (2 * AT) * 4 + lloc;

  v8f acc[4] = {};
  const int NCH = H_ / KC;
  async_A(gA, voff, lA_l, 32u * H_ * 4u);
  async_B(gB, voff, lB_l, 32u * H_ * 4u);
  for (int c = 0; c < NCH; ++c) {
    int cur = c & 1;
    if (c + 1 < NCH) {
      int nxt = (c + 1) & 1;
      async_A(gA + (c + 1) * KC, voff, lA_l + nxt * AT * 4, 32u * H_ * 4u);
      async_B(gB + (c + 1) * KC, voff, lB_l + nxt * BT * 4, 32u * H_ * 4u);
      ASYNC_WAIT(6);
    } else {
      ASYNC_WAIT(0);
    }
    __syncthreads();
    mma_chunk<1>(lds + cur * AT, lds + cur * AT, lds + 2 * AT + cur * BT, acc, acc);
    __syncthreads();
  }
  int lane = threadIdx.x & 31, wv = threadIdx.x >> 5;
  int half = lane >> 4, lm = lane & 15;
  int gmBase = m0 + wv * 16 + half * 8;
  float* ob = outp + (size_t)b * H_ * S_;
#pragma unroll
  for (int j = 0; j < 4; ++j)
#pragma unroll
    for (int r = 0; r < 8; ++r)
      ob[(size_t)(gmBase + r) * S_ + n0 + j * 16 + lm] = acc[j][r];
}

// rel[b,k] = 0.5*sigmoid(Wg[k]@h+bg[k]) * (W[k]@h+bi[k]); h from hsrc [B,S,H]; rel: [B,K,H,S]
__global__ __launch_bounds__(256) void k_gemm_gated(const float* __restrict__ Wg,
                                                    const float* __restrict__ bg,
                                                    const float* __restrict__ W,
                                                    const float* __restrict__ bi,
                                                    const float* __restrict__ hsrc,
                                                    float* __restrict__ rel) {
  __shared__ __align__(128) float lds[4 * AT + 2 * BT];   // Ag x2, Aw x2, B x2
  int bk = blockIdx.x;
  int b = bk >> 2, k = bk & 3;
  int m0 = blockIdx.y * MBLK;
  int n0 = blockIdx.z * NBLK;
  const float* gAg = Wg + (size_t)k * H_ * H_ + (size_t)m0 * H_;
  const float* gAw = W + (size_t)k * H_ * H_ + (size_t)m0 * H_;
  const float* gB  = hsrc + (size_t)b * S_ * H_ + (size_t)n0 * H_;

  unsigned t = threadIdx.x;
  unsigned voff = (((t >> 3) * H_) + (t & 7) * 4) * 4;
  unsigned lloc = ((t >> 3) * ASTR + (t & 7) * 4) * 4;
  unsigned base = lds_addr(lds);
  unsigned lAg_l = base + lloc;
  unsigned lAw_l = lAg_l + (2 * AT) * 4;
  unsigned lB_l  = base + (4 * AT) * 4 + lloc;

  v8f accG[4] = {}, accW[4] = {};
  const int NCH = H_ / KC;
  async_A(gAg, voff, lAg_l, 32u * H_ * 4u);
  async_A(gAw, voff, lAw_l, 32u * H_ * 4u);
  async_B(gB, voff, lB_l, 32u * H_ * 4u);
  for (int c = 0; c < NCH; ++c) {
    int cur = c & 1;
    if (c + 1 < NCH) {
      int nxt = (c + 1) & 1;
      async_A(gAg + (c + 1) * KC, voff, lAg_l + nxt * AT * 4, 32u * H_ * 4u);
      async_A(gAw + (c + 1) * KC, voff, lAw_l + nxt * AT * 4, 32u * H_ * 4u);
      async_B(gB + (c + 1) * KC, voff, lB_l + nxt * BT * 4, 32u * H_ * 4u);
      ASYNC_WAIT(10);
    } else {
      ASYNC_WAIT(0);
    }
    __syncthreads();
    mma_chunk<2>(lds + cur * AT, lds + 2 * AT + cur * AT, lds + 4 * AT + cur * BT, accG, accW);
    __syncthreads();
  }
  int lane = threadIdx.x & 31, wv = threadIdx.x >> 5;
  int half = lane >> 4, lm = lane & 15;
  int gmBase = m0 + wv * 16 + half * 8;
  const float* bgp = bg + (size_t)k * H_;
  const float* bp  = bi + (size_t)k * H_;
  float bgv[8], bv[8];
#pragma unroll
  for (int r = 0; r < 8; ++r) { bgv[r] = bgp[gmBase + r]; bv[r] = bp[gmBase + r]; }
  float* rp = rel + ((size_t)bk * H_) * S_;
#pragma unroll
  for (int j = 0; j < 4; ++j)
#pragma unroll
    for (int r = 0; r < 8; ++r) {
      float g = 1.0f / (1.0f + __expf(-(accG[j][r] + bgv[r])));
      float v = 0.5f * g * (accW[j][r] + bv[r]);
      rp[(size_t)(gmBase + r) * S_ + n0 + j * 16 + lm] = v;
    }
}

// dst = accbuf + sum_k rel[b,k] @ Badj[b,k]; Badj supplied k-contiguous (bsrc[bk][t][s]).
// storeTrans=0: dst[B,H,S] (acc layout); storeTrans=1: dst[B,S,H] (hT / final output layout)
__global__ __launch_bounds__(256) void k_gemm_adj(const float* __restrict__ rel,
                                                  const float* __restrict__ bsrc,
                                                  const float* __restrict__ accbuf,
                                                  float* __restrict__ dst,
                                                  int doRelu, int storeTrans) {
  __shared__ __align__(128) float lds[2 * AT + 2 * BT];
  int b  = blockIdx.x;
  int m0 = blockIdx.y * MBLK;
  int t0 = blockIdx.z * NBLK;

  unsigned t = threadIdx.x;
  unsigned voffA = (((t >> 3) * S_) + (t & 7) * 4) * 4;
  unsigned lloc  = ((t >> 3) * ASTR + (t & 7) * 4) * 4;
  unsigned base = lds_addr(lds);
  unsigned lA_l = base + lloc;
  unsigned lB_l = base + (2 * AT) * 4 + lloc;

  v8f acc[4] = {};
  const int NCH = K_ * (S_ / KC);   // 16 chunks: k = it>>2, kc = (it&3)*KC
  {
    const float* rp = rel + ((size_t)(b * K_) * H_) * S_;
    const float* ap = bsrc + ((size_t)(b * K_) * S_) * S_;
    async_A(rp + (size_t)m0 * S_, voffA, lA_l, 32u * S_ * 4u);
    async_B(ap + (size_t)t0 * S_, voffA, lB_l, 32u * S_ * 4u);
  }
  for (int it = 0; it < NCH; ++it) {
    int cur = it & 1;
    if (it + 1 < NCH) {
      int nxt = (it + 1) & 1;
      int k2 = (it + 1) >> 2, kc2 = ((it + 1) & 3) * KC;
      const float* rp = rel + ((size_t)(b * K_ + k2) * H_) * S_;
      const float* ap = bsrc + ((size_t)(b * K_ + k2) * S_) * S_;
      async_A(rp + (size_t)m0 * S_ + kc2, voffA, lA_l + nxt * AT * 4, 32u * S_ * 4u);
      async_B(ap + (size_t)t0 * S_ + kc2, voffA, lB_l + nxt * BT * 4, 32u * S_ * 4u);
      ASYNC_WAIT(6);
    } else {
      ASYNC_WAIT(0);
    }
    __syncthreads();
    mma_chunk<1>(lds + cur * AT, lds + cur * AT, lds + 2 * AT + cur * BT, acc, acc);
    __syncthreads();
  }
  int lane = threadIdx.x & 31, wv = threadIdx.x >> 5;
  int half = lane >> 4, lm = lane & 15;
  int gmBase = m0 + wv * 16 + half * 8;
#pragma unroll
  for (int j = 0; j < 4; ++j)
#pragma unroll
    for (int r = 0; r < 8; ++r) {
      int gm = gmBase + r;
      int gn = t0 + j * 16 + lm;
      float v = accbuf[((size_t)b * H_ + gm) * S_ + gn] + acc[j][r];
      if (doRelu) v = fmaxf(v, 0.0f);
      if (storeTrans) dst[((size_t)b * S_ + gn) * H_ + gm] = v;
      else            dst[((size_t)b * H_ + gm) * S_ + gn] = v;
    }
}

extern "C" void kernel_launch(void* const* d_in, const int* in_sizes, int n_in,
                              void* d_out, int out_size, void* d_ws, size_t ws_size,
                              hipStream_t stream) {
  const float* sent   = (const float*)d_in[0];   // [B,S,H] == hT layout
  const float* adj    = (const float*)d_in[1];
  const float* W_in   = (const float*)d_in[2];
  const float* b_in   = (const float*)d_in[3];
  const float* W_out  = (const float*)d_in[4];
  const float* b_out  = (const float*)d_in[5];
  const float* Wg_in  = (const float*)d_in[6];
  const float* bg_in  = (const float*)d_in[7];
  const float* Wg_out = (const float*)d_in[8];
  const float* bg_out = (const float*)d_in[9];
  const float* W_loop = (const float*)d_in[10];
  float* out = (float*)d_out;

  float* hT   = (float*)d_ws;                       // B*S*H
  float* acc  = hT + (size_t)B_ * S_ * H_;          // B*H*S
  float* rel  = acc + (size_t)B_ * H_ * S_;         // B*K*H*S
  float* adjT = rel + (size_t)B_ * K_ * H_ * S_;    // B*K*S*S

  k_transpose_adj<<<dim3(S_ / 32, S_ / 32, B_ * K_), 256, 0, stream>>>(adj, adjT);

  for (int l = 0; l < L_; ++l) {
    size_t wOff = (size_t)l * K_ * H_ * H_;
    size_t bOff = (size_t)l * K_ * H_;
    const float* hsrc = (l == 0) ? sent : hT;
    int last = (l == L_ - 1);

    k_gemm_plain<<<dim3(B_, H_ / MBLK, S_ / NBLK), 256, 0, stream>>>(
        W_loop + (size_t)l * H_ * H_, hsrc, acc);
    // in direction: B = adj  -> k-contiguous source is adjT
    k_gemm_gated<<<dim3(B_ * K_, H_ / MBLK, S_ / NBLK), 256, 0, stream>>>(
        Wg_in + wOff, bg_in + bOff, W_in + wOff, b_in + bOff, hsrc, rel);
    k_gemm_adj<<<dim3(B_, H_ / MBLK, S_ / NBLK), 256, 0, stream>>>(
        rel, adjT, acc, acc, /*relu=*/0, /*storeTrans=*/0);
    // out direction: B = adj^T -> k-contiguous source is adj itself
    k_gemm_gated<<<dim3(B_ * K_, H_ / MBLK, S_ / NBLK), 256, 0, stream>>>(
        Wg_out + wOff, bg_out + bOff, W_out + wOff, b_out + bOff, hsrc, rel);
    k_gemm_adj<<<dim3(B_, H_ / MBLK, S_ / NBLK), 256, 0, stream>>>(
        rel, adj, acc, last ? out : hT, /*relu=*/1, /*storeTrans=*/1);
  }
}